// VectorQuantizerEMA_24352464568640
// MI455X (gfx1250) — compile-verified
//
#include <hip/hip_runtime.h>
#include <hip/hip_bf16.h>
#include <math.h>

#define NTOT 163840      // 512*320 flattened vectors
#define DIMD 256         // embedding dim
#define KCB  1024        // codebook size
#define DECAYF 0.99f
#define ONEMD  0.01f
#define PGRP 96          // dword stride per d-pair group (96 % 64 == 32 -> conflict-free b64 loads)

typedef __attribute__((ext_vector_type(2))) float v2f;
typedef __attribute__((ext_vector_type(4))) float v4f;
typedef __attribute__((ext_vector_type(8))) float v8f;

// ---------------- zero fill ----------------
__global__ void vq_zero_f(float* __restrict__ p, int count) {
    int i = blockIdx.x * blockDim.x + threadIdx.x;
    int stride = gridDim.x * blockDim.x;
    for (; i < count; i += stride) p[i] = 0.0f;
}

__global__ void vq_zero_f4(v4f* __restrict__ p, int count4) {
    int i = blockIdx.x * blockDim.x + threadIdx.x;
    int stride = gridDim.x * blockDim.x;
    v4f z = {0.f, 0.f, 0.f, 0.f};
    for (; i < count4; i += stride) __builtin_nontemporal_store(z, p + i);
}

// ---------------- ||x_n||^2 (column of [D,N] input, coalesced) ----------------
__global__ __launch_bounds__(256) void vq_xnorm_kernel(const float* __restrict__ IN,
                                                       float* __restrict__ xs) {
    int n = blockIdx.x * 256 + threadIdx.x;
    float s = 0.0f;
    for (int d = 0; d < DIMD; ++d) {
        float v = IN[(size_t)d * NTOT + n];
        s += v * v;
    }
    xs[n] = s;
}

// ---------------- ||e_k||^2 (wave per row, wave32 reduce) ----------------
__global__ __launch_bounds__(256) void vq_enorm_kernel(const float* __restrict__ E,
                                                       float* __restrict__ es) {
    int wave = threadIdx.x >> 5;
    int lane = threadIdx.x & 31;
    int k = blockIdx.x * 8 + wave;
    float s = 0.0f;
    for (int i = 0; i < DIMD / 32; ++i) {
        float v = E[(size_t)k * DIMD + lane + 32 * i];
        s += v * v;
    }
    for (int off = 16; off >= 1; off >>= 1) s += __shfl_xor(s, off, 32);
    if (lane == 0) es[k] = s;
}

// ---------------- main WMMA distance + argmin kernel ----------------
// block = 256 threads = 8 waves; block owns 32 rows of flat X (two 16-row M tiles).
// Each wave: two accumulators sharing one B fragment -> half the global B loads per WMMA.
// A-tile stored d-pair interleaved so each A fragment is a single conflict-free ds_load_b64.
// WMMA f32 16x16x4:  M = n (rows), N = codes, K = d.
__global__ __launch_bounds__(256) void vq_dist_kernel(
    const float* __restrict__ IN,    // [D, N] (flat X is its transpose)
    const float* __restrict__ E,     // [K, D]
    const float* __restrict__ xs,    // [N]
    const float* __restrict__ es,    // [K]
    float* __restrict__ dist,        // [N, K] output
    float* __restrict__ oidxf,       // [N] index output (as float)
    int* __restrict__ idx,           // [N] scratch int indices
    float* __restrict__ cnt)         // [K] histogram
{
    // element (d, c) lives at dword (d>>1)*PGRP + 2c + (d&1)
    __shared__ float Atile[(DIMD / 2) * PGRP];   // 48 KB
    __shared__ float candv[8][32];
    __shared__ int   candi[8][32];

    const int n0   = blockIdx.x * 32;
    const int tid  = threadIdx.x;
    const int wave = tid >> 5;
    const int lane = tid & 31;
    const int m    = lane & 15;      // column-in-tile (A: M row, B/D: N col)
    const int half = lane >> 4;      // K-pair / M+8 selector
    const int ka   = half * 2;       // K-pair offset inside 16x16x4 fragment

    // stage A tile: b64 stores, lanes cover banks {2c, 2c+1} -> all 64 banks
    for (int e = tid; e < (DIMD / 2) * 32; e += 256) {
        int p = e >> 5, c = e & 31;
        v2f val;
        val.x = IN[(size_t)(2 * p) * NTOT + n0 + c];
        val.y = IN[(size_t)(2 * p + 1) * NTOT + n0 + c];
        *(v2f*)&Atile[p * PGRP + 2 * c] = val;
    }
    __syncthreads();

    float xsh0[8], xsh1[8];
#pragma unroll
    for (int r = 0; r < 8; ++r) {
        xsh0[r] = xs[n0 + r + 8 * half];
        xsh1[r] = xs[n0 + 16 + r + 8 * half];
    }

    float bv0[8], bv1[8];
    int   bi0[8], bi1[8];
#pragma unroll
    for (int r = 0; r < 8; ++r) {
        bv0[r] = 3.4e38f; bi0[r] = 0;
        bv1[r] = 3.4e38f; bi1[r] = 0;
    }

    // per-lane A base: d-pair group (d0/2 + half), column m (a1: +16 columns = +32 dwords)
    const float* abase = &Atile[half * PGRP + 2 * m];

    for (int pass = 0; pass < 8; ++pass) {
        const int c0 = pass * 128 + wave * 16;    // 16-code block for this wave
        v8f acc0 = {0.f, 0.f, 0.f, 0.f, 0.f, 0.f, 0.f, 0.f};
        v8f acc1 = {0.f, 0.f, 0.f, 0.f, 0.f, 0.f, 0.f, 0.f};
        const float* brow = E + (size_t)(c0 + m) * DIMD + ka;  // 8B aligned
#pragma unroll 4
        for (int d0 = 0; d0 < DIMD; d0 += 4) {
            // B 4x16 f32 layout: lane m = column c0+m; K pair selected by half
            v2f b = *(const v2f*)(brow + d0);
            // A fragments: single b64 LDS loads, even-aligned pairs feed WMMA directly
            const float* ap = abase + (d0 >> 1) * PGRP;
            v2f a0 = *(const v2f*)ap;
            v2f a1 = *(const v2f*)(ap + 32);
            acc0 = __builtin_amdgcn_wmma_f32_16x16x4_f32(
                false, a0, false, b, (short)0, acc0, false, false);
            acc1 = __builtin_amdgcn_wmma_f32_16x16x4_f32(
                false, a1, false, b, (short)0, acc1, false, false);
        }
        const float ev   = es[c0 + m];
        const int   cidx = c0 + m;
#pragma unroll
        for (int r = 0; r < 8; ++r) {
            // D layout: VGPR r, lane -> row base + r + 8*half, col c0+m
            float dv0 = xsh0[r] + ev - 2.0f * acc0[r];
            __builtin_nontemporal_store(
                dv0, dist + (size_t)(n0 + r + 8 * half) * KCB + cidx);
            if (dv0 < bv0[r]) { bv0[r] = dv0; bi0[r] = cidx; }
            float dv1 = xsh1[r] + ev - 2.0f * acc1[r];
            __builtin_nontemporal_store(
                dv1, dist + (size_t)(n0 + 16 + r + 8 * half) * KCB + cidx);
            if (dv1 < bv1[r]) { bv1[r] = dv1; bi1[r] = cidx; }
        }
    }

    // argmin across the 16 code lanes of each half (tie-break: lowest index)
#pragma unroll
    for (int r = 0; r < 8; ++r) {
        float v = bv0[r]; int i = bi0[r];
        float w = bv1[r]; int j = bi1[r];
        for (int off = 8; off >= 1; off >>= 1) {
            float ov = __shfl_xor(v, off, 16);
            int   oi = __shfl_xor(i, off, 16);
            if (ov < v || (ov == v && oi < i)) { v = ov; i = oi; }
            float ow = __shfl_xor(w, off, 16);
            int   oj = __shfl_xor(j, off, 16);
            if (ow < w || (ow == w && oj < j)) { w = ow; j = oj; }
        }
        if (m == 0) {
            candv[wave][r + 8 * half]      = v;
            candi[wave][r + 8 * half]      = i;
            candv[wave][16 + r + 8 * half] = w;
            candi[wave][16 + r + 8 * half] = j;
        }
    }
    __syncthreads();

    // combine the 8 waves' candidates (each covered disjoint code ranges)
    if (tid < 32) {
        float v = candv[0][tid]; int i = candi[0][tid];
        for (int w = 1; w < 8; ++w) {
            float ov = candv[w][tid]; int oi = candi[w][tid];
            if (ov < v || (ov == v && oi < i)) { v = ov; i = oi; }
        }
        idx[n0 + tid]   = i;
        oidxf[n0 + tid] = (float)i;
        unsafeAtomicAdd(&cnt[i], 1.0f);
    }
}

// ---------------- one-hot scatter (region pre-zeroed) ----------------
__global__ __launch_bounds__(256) void vq_onehot_kernel(const int* __restrict__ idx,
                                                        float* __restrict__ enc) {
    int n = blockIdx.x * 256 + threadIdx.x;
    enc[(size_t)n * KCB + idx[n]] = 1.0f;
}

// ---------------- dw = encodings^T @ X via scatter-add ----------------
__global__ __launch_bounds__(256) void vq_dw_kernel(const float* __restrict__ IN,
                                                    const int* __restrict__ idx,
                                                    float* __restrict__ dw) {
    int n = blockIdx.x * 256 + threadIdx.x;
    int k = idx[n];
    float* row = dw + (size_t)k * DIMD;
    for (int d = 0; d < DIMD; ++d) {
        unsafeAtomicAdd(&row[d], IN[(size_t)d * NTOT + n]);  // coalesced read
    }
}

// ---------------- EMA cluster-size + perplexity (single block, 1024 thr) ----
__global__ __launch_bounds__(1024) void vq_ema_stats_kernel(
    const float* __restrict__ ecs, const float* __restrict__ cnt,
    float* __restrict__ csf, float* __restrict__ o_cs, float* __restrict__ o_perp) {
    __shared__ float red[32];
    int t = threadIdx.x, lane = t & 31, w = t >> 5;

    float csr = ecs[t] * DECAYF + ONEMD * cnt[t];
    float s = csr;
    for (int off = 16; off >= 1; off >>= 1) s += __shfl_xor(s, off, 32);
    if (lane == 0) red[w] = s;
    __syncthreads();
    if (t < 32) {
        float v = red[t];
        for (int off = 16; off >= 1; off >>= 1) v += __shfl_xor(v, off, 32);
        if (t == 0) red[0] = v;
    }
    __syncthreads();
    float ntot = red[0];
    __syncthreads();

    float cf = (csr + 1e-5f) / (ntot + (float)KCB * 1e-5f) * ntot;
    csf[t]  = cf;
    o_cs[t] = cf;

    float avg = cnt[t] * (1.0f / (float)NTOT);
    float pl  = avg * logf(avg + 1e-10f);
    s = pl;
    for (int off = 16; off >= 1; off >>= 1) s += __shfl_xor(s, off, 32);
    if (lane == 0) red[w] = s;
    __syncthreads();
    if (t < 32) {
        float v = red[t];
        for (int off = 16; off >= 1; off >>= 1) v += __shfl_xor(v, off, 32);
        if (t == 0) o_perp[0] = expf(-v);
    }
}

// ---------------- ema_w_new and embedding_new ----------------
__global__ __launch_bounds__(256) void vq_emaw_kernel(const float* __restrict__ ew,
                                                      const float* __restrict__ dw,
                                                      const float* __restrict__ csf,
                                                      float* __restrict__ o_ew,
                                                      float* __restrict__ o_emb) {
    int i = blockIdx.x * 256 + threadIdx.x;   // grid = K*D/256
    float w = ew[i] * DECAYF + ONEMD * dw[i];
    o_ew[i]  = w;
    o_emb[i] = w / csf[i >> 8];
}

// ---------------- quantize (gather) + commitment loss ----------------
__global__ __launch_bounds__(256) void vq_quant_kernel(const float* __restrict__ IN,
                                                       const float* __restrict__ emb,
                                                       const int* __restrict__ idx,
                                                       float* __restrict__ oq,
                                                       float* __restrict__ loss) {
    __shared__ float red[8];
    size_t i = (size_t)blockIdx.x * 256 + threadIdx.x;   // i = d*N + n
    int d = (int)(i / NTOT);
    int n = (int)(i % NTOT);
    float q = emb[(size_t)idx[n] * DIMD + d];
    float x = IN[i];
    __builtin_nontemporal_store(q, oq + i);
    float df = q - x;
    float s = df * df;
    for (int off = 16; off >= 1; off >>= 1) s += __shfl_xor(s, off, 32);
    int lane = threadIdx.x & 31, w = threadIdx.x >> 5;
    if (lane == 0) red[w] = s;
    __syncthreads();
    if (threadIdx.x == 0) {
        float v = 0.0f;
        for (int j = 0; j < 8; ++j) v += red[j];
        unsafeAtomicAdd(loss, v);
    }
}

__global__ void vq_final_kernel(const float* __restrict__ loss, float* __restrict__ o0) {
    o0[0] = 0.25f * loss[0] / (float)((size_t)DIMD * NTOT);
}

// ---------------- host ----------------
extern "C" void kernel_launch(void* const* d_in, const int* in_sizes, int n_in,
                              void* d_out, int out_size, void* d_ws, size_t ws_size,
                              hipStream_t stream) {
    (void)in_sizes; (void)n_in; (void)out_size; (void)ws_size;

    const float* IN  = (const float*)d_in[0];   // [256, 512, 320]
    const float* E   = (const float*)d_in[1];   // [1024, 256]
    const float* ECS = (const float*)d_in[2];   // [1024]
    const float* EW  = (const float*)d_in[3];   // [1024, 256]

    float* out    = (float*)d_out;
    float* o_loss = out;                                   // 1
    float* o_q    = o_loss + 1;                            // D*N
    float* o_perp = o_q + (size_t)DIMD * NTOT;             // 1
    float* o_enc  = o_perp + 1;                            // N*K
    float* o_dist = o_enc + (size_t)NTOT * KCB;            // N*K
    float* o_idx  = o_dist + (size_t)NTOT * KCB;           // N
    float* o_emb  = o_idx + NTOT;                          // K*D
    float* o_cs   = o_emb + (size_t)KCB * DIMD;            // K
    float* o_ew   = o_cs + KCB;                            // K*D

    float* xs   = (float*)d_ws;                 // N
    float* es   = xs + NTOT;                    // K
    float* cnt  = es + KCB;                     // K     (zeroed)
    float* csf  = cnt + KCB;                    // K     (zeroed, rewritten)
    float* dw   = csf + KCB;                    // K*D   (zeroed)
    float* loss = dw + (size_t)KCB * DIMD;      // 1     (zeroed)
    int*   idx  = (int*)(loss + 1);             // N ints

    // 1) zero accumulators (cnt, csf, dw, loss are contiguous)
    vq_zero_f<<<256, 256, 0, stream>>>(cnt, KCB + KCB + KCB * DIMD + 1);
    // 2) zero the one-hot encodings region (N*K floats)
    vq_zero_f4<<<8192, 256, 0, stream>>>((v4f*)o_enc, (int)((size_t)NTOT * KCB / 4));
    // 3) norms
    vq_xnorm_kernel<<<NTOT / 256, 256, 0, stream>>>(IN, xs);
    vq_enorm_kernel<<<KCB / 8, 256, 0, stream>>>(E, es);
    // 4) WMMA distances + argmin + histogram (32 rows per block)
    vq_dist_kernel<<<NTOT / 32, 256, 0, stream>>>(IN, E, xs, es, o_dist, o_idx, idx, cnt);
    // 5) one-hot
    vq_onehot_kernel<<<NTOT / 256, 256, 0, stream>>>(idx, o_enc);
    // 6) dw scatter-add
    vq_dw_kernel<<<NTOT / 256, 256, 0, stream>>>(IN, idx, dw);
    // 7) EMA cluster size + perplexity
    vq_ema_stats_kernel<<<1, 1024, 0, stream>>>(ECS, cnt, csf, o_cs, o_perp);
    // 8) ema_w_new + embedding_new
    vq_emaw_kernel<<<(KCB * DIMD) / 256, 256, 0, stream>>>(EW, dw, csf, o_ew, o_emb);
    // 9) quantize gather + loss
    vq_quant_kernel<<<(int)(((size_t)DIMD * NTOT) / 256), 256, 0, stream>>>(IN, o_emb, idx, o_q, loss);
    // 10) scalar finalize
    vq_final_kernel<<<1, 1, 0, stream>>>(loss, o_loss);
}